// AutoEncoder_80977313399099
// MI455X (gfx1250) — compile-verified
//
#include <hip/hip_runtime.h>
#include <hip/hip_bf16.h>

// ---------------------------------------------------------------------------
// VQ-VAE forward for MI455X (gfx1250).
//   N=256, x: 1x128x128
//   enc: conv4x4s2(1->16)+BN+ReLU -> conv4x4s2(16->32)+BN+ReLU -> 1x1(32->64)+BN
//   VQ : argmin_k ||z - e_k||^2 over K=512, C=64  (WMMA f32 16x16x4 GEMM)
//   dec: 1x1(64->32)+BN+ReLU -> convT4x4s2(32->16)+BN+ReLU -> convT4x4s2(16->1)+sigmoid
// Outputs concat: x_tilde (4194304) | z_e_x (16777216) | z_q_x (16777216)
// ---------------------------------------------------------------------------

typedef float v2f __attribute__((ext_vector_type(2)));
typedef float v8f __attribute__((ext_vector_type(8)));

#define NBATCH 256
#define KCODES 512
#define CDIM   64
#define MPOS   (NBATCH * 32 * 32)   // 262144 spatial positions at latent res

// ---------------- conv1: 1->16, k4 s2 p1, 128x128 -> 64x64 ------------------
__global__ void conv1_k(const float* __restrict__ x, const float* __restrict__ w,
                        const float* __restrict__ bias, float* __restrict__ y) {
    int idx = blockIdx.x * blockDim.x + threadIdx.x;           // 16,777,216
    int ow = idx & 63, oh = (idx >> 6) & 63, co = (idx >> 12) & 15, n = idx >> 16;
    const float* xp = x + (long)n * 128 * 128;
    const float* wp = w + co * 16;
    float acc = bias[co];
    int ih0 = oh * 2 - 1, iw0 = ow * 2 - 1;
#pragma unroll
    for (int kh = 0; kh < 4; ++kh) {
        int ih = ih0 + kh; if ((unsigned)ih >= 128u) continue;
#pragma unroll
        for (int kw = 0; kw < 4; ++kw) {
            int iw = iw0 + kw; if ((unsigned)iw >= 128u) continue;
            acc += xp[ih * 128 + iw] * wp[kh * 4 + kw];
        }
    }
    y[idx] = acc;
}

// ---------------- conv2: 16->32, k4 s2 p1, 64x64 -> 32x32 -------------------
__global__ void conv2_k(const float* __restrict__ x, const float* __restrict__ w,
                        const float* __restrict__ bias, float* __restrict__ y) {
    int idx = blockIdx.x * blockDim.x + threadIdx.x;           // 8,388,608
    int ow = idx & 31, oh = (idx >> 5) & 31, co = (idx >> 10) & 31, n = idx >> 15;
    const float* xn = x + (long)n * 16 * 4096;
    const float* wp = w + co * 256;
    float acc = bias[co];
    int ih0 = oh * 2 - 1, iw0 = ow * 2 - 1;
    for (int ci = 0; ci < 16; ++ci) {
        const float* xc = xn + ci * 4096;
        const float* wc = wp + ci * 16;
#pragma unroll
        for (int kh = 0; kh < 4; ++kh) {
            int ih = ih0 + kh; if ((unsigned)ih >= 64u) continue;
#pragma unroll
            for (int kw = 0; kw < 4; ++kw) {
                int iw = iw0 + kw; if ((unsigned)iw >= 64u) continue;
                acc += xc[ih * 64 + iw] * wc[kh * 4 + kw];
            }
        }
    }
    y[idx] = acc;
}

// ---------------- generic 1x1 conv ------------------------------------------
__global__ void conv1x1_k(const float* __restrict__ x, const float* __restrict__ w,
                          const float* __restrict__ bias, float* __restrict__ y,
                          int Cin, int Cout, int HW) {
    int idx = blockIdx.x * blockDim.x + threadIdx.x;
    int hw = idx % HW;
    int co = (idx / HW) % Cout;
    int n  = idx / (HW * Cout);
    const float* xn = x + ((long)n * Cin) * HW + hw;
    const float* wp = w + co * Cin;
    float acc = bias[co];
    for (int ci = 0; ci < Cin; ++ci) acc += xn[(long)ci * HW] * wp[ci];
    y[idx] = acc;
}

// ---------------- transposed conv k4 s2 p1 (torch weight [Cin,Cout,4,4]) ----
__global__ void convT_k(const float* __restrict__ x, const float* __restrict__ w,
                        const float* __restrict__ bias, float* __restrict__ y,
                        int Cin, int Cout, int Hin, int Hout, int sig) {
    int idx = blockIdx.x * blockDim.x + threadIdx.x;  // N*Cout*Hout*Hout
    int ow = idx % Hout;
    int oh = (idx / Hout) % Hout;
    int co = (idx / (Hout * Hout)) % Cout;
    int n  = idx / (Hout * Hout * Cout);
    const float* xn = x + (long)n * Cin * Hin * Hin;
    float acc = bias[co];
    int kh0 = (oh + 1) & 1, kw0 = (ow + 1) & 1;
#pragma unroll
    for (int dh = 0; dh < 2; ++dh) {
        int kh = kh0 + 2 * dh;
        int ih2 = oh + 1 - kh; if (ih2 < 0) continue;
        int ih = ih2 >> 1;     if (ih >= Hin) continue;
#pragma unroll
        for (int dw = 0; dw < 2; ++dw) {
            int kw = kw0 + 2 * dw;
            int iw2 = ow + 1 - kw; if (iw2 < 0) continue;
            int iw = iw2 >> 1;     if (iw >= Hin) continue;
            for (int ci = 0; ci < Cin; ++ci)
                acc += xn[(ci * Hin + ih) * Hin + iw] * w[((ci * Cout + co) * 4 + kh) * 4 + kw];
        }
    }
    if (sig) acc = 1.0f / (1.0f + __expf(-acc));
    y[idx] = acc;
}

// ---------------- BN batch stats (deterministic 2-stage reduction) ----------
#define STATS_NB 64
__global__ void stats_partial_k(const float* __restrict__ x, float* __restrict__ part,
                                int C, int HW, int N) {
    int c = blockIdx.x, b = blockIdx.y;
    long L = (long)N * HW;
    float s = 0.f, sq = 0.f;
    for (long j = (long)b * blockDim.x + threadIdx.x; j < L; j += (long)STATS_NB * blockDim.x) {
        long n = j / HW, i = j - n * HW;
        float v = x[(n * C + c) * (long)HW + i];
        s += v; sq += v * v;
    }
    __shared__ float ss[256], sr[256];
    ss[threadIdx.x] = s; sr[threadIdx.x] = sq;
    __syncthreads();
    for (int st = blockDim.x >> 1; st > 0; st >>= 1) {
        if ((int)threadIdx.x < st) {
            ss[threadIdx.x] += ss[threadIdx.x + st];
            sr[threadIdx.x] += sr[threadIdx.x + st];
        }
        __syncthreads();
    }
    if (threadIdx.x == 0) {
        part[(c * STATS_NB + b) * 2 + 0] = ss[0];
        part[(c * STATS_NB + b) * 2 + 1] = sr[0];
    }
}

__global__ void stats_finalize_k(const float* __restrict__ part,
                                 const float* __restrict__ g, const float* __restrict__ beta,
                                 float* __restrict__ scsh, int C, float invL) {
    int c = threadIdx.x;
    if (c >= C) return;
    float s = 0.f, sq = 0.f;
    for (int b = 0; b < STATS_NB; ++b) {
        s  += part[(c * STATS_NB + b) * 2 + 0];
        sq += part[(c * STATS_NB + b) * 2 + 1];
    }
    float mean = s * invL;
    float var  = sq * invL - mean * mean;
    float sc   = g[c] * rsqrtf(var + 1e-5f);
    scsh[c]     = sc;
    scsh[C + c] = beta[c] - mean * sc;
}

__global__ void bn_apply_k(const float* __restrict__ x, float* __restrict__ y,
                           const float* __restrict__ scsh, int C, int HW, int relu) {
    int idx = blockIdx.x * blockDim.x + threadIdx.x;
    int c = (idx / HW) % C;
    float v = scsh[c] * x[idx] + scsh[C + c];
    if (relu) v = fmaxf(v, 0.f);
    y[idx] = v;
}

// ---------------- VQ argmin via V_WMMA_F32_16X16X4_F32 ----------------------
// score(m,k) = ||e_k||^2 - 2 * dot(z_m, e_k)   (z.z constant per row)
// One wave per 16-row M tile; 4 waves per block; codebook staged in LDS.
#define VQ_WAVES 4
__global__ void vq_argmin_k(const float* __restrict__ ze,   // [N,64,32,32]
                            const float* __restrict__ emb,  // [512,64]
                            int* __restrict__ lat) {
    extern __shared__ float smem[];
    float* sE = smem;                 // 512*64 codebook
    float* sC = smem + KCODES * CDIM; // 512 squared norms

    for (int i = threadIdx.x; i < KCODES * CDIM; i += blockDim.x) sE[i] = emb[i];
    __syncthreads();
    for (int k = threadIdx.x; k < KCODES; k += blockDim.x) {
        float s = 0.f;
        for (int c = 0; c < CDIM; ++c) { float v = sE[k * CDIM + c]; s += v * v; }
        sC[k] = s;
    }
    __syncthreads();

    int lane  = threadIdx.x & 31;
    int wave  = threadIdx.x >> 5;
    int m0    = (blockIdx.x * VQ_WAVES + wave) * 16;
    int lrow  = lane & 15;      // row within tile held by this lane (A side)
    int khalf = lane >> 4;      // lanes 0-15: K offsets {0,1}; 16-31: {2,3}

    // A fragments: 16 k-steps of K=4, kept resident across all N tiles.
    int row = m0 + lrow;
    int n   = row >> 10;
    int hw  = row & 1023;
    const float* zrow = ze + (long)n * CDIM * 1024 + hw;  // channel stride 1024
    v2f a[16];
#pragma unroll
    for (int ks = 0; ks < 16; ++ks) {
        int c0 = ks * 4 + khalf * 2;
        a[ks].x = zrow[(long)c0 * 1024];
        a[ks].y = zrow[(long)(c0 + 1) * 1024];
    }

    float best[8]; int bidx[8];
#pragma unroll
    for (int j = 0; j < 8; ++j) { best[j] = 3.4e38f; bidx[j] = 0; }

    int col16 = lane & 15;  // column within N tile for B/C/D fragments
    for (int t = 0; t < KCODES / 16; ++t) {
        int n0 = t * 16;
        int col = n0 + col16;
        v8f acc = {0.f, 0.f, 0.f, 0.f, 0.f, 0.f, 0.f, 0.f};
#pragma unroll
        for (int ks = 0; ks < 16; ++ks) {
            int c0 = ks * 4 + khalf * 2;
            v2f b;
            b.x = sE[col * CDIM + c0];
            b.y = sE[col * CDIM + c0 + 1];
            acc = __builtin_amdgcn_wmma_f32_16x16x4_f32(
                false, a[ks], false, b, (short)0, acc, false, false);
        }
        float cn = sC[col];
#pragma unroll
        for (int j = 0; j < 8; ++j) {
            float score = cn - 2.0f * acc[j];
            if (score < best[j]) { best[j] = score; bidx[j] = col; }
        }
    }

    // Reduce over the 16 lanes that share each row (butterfly within half-wave).
#pragma unroll
    for (int j = 0; j < 8; ++j) {
        for (int off = 8; off > 0; off >>= 1) {
            float ob = __shfl_xor(best[j], off, 16);
            int   oi = __shfl_xor(bidx[j], off, 16);
            if (ob < best[j] || (ob == best[j] && oi < bidx[j])) { best[j] = ob; bidx[j] = oi; }
        }
    }
    if (lane == 0) {
#pragma unroll
        for (int j = 0; j < 8; ++j) lat[m0 + j] = bidx[j];
    }
    if (lane == 16) {
#pragma unroll
        for (int j = 0; j < 8; ++j) lat[m0 + 8 + j] = bidx[j];
    }
}

// ---------------- z_q gather: zq[n,c,h,w] = emb[lat[m], c] ------------------
__global__ void scatter_zq_k(const int* __restrict__ lat, const float* __restrict__ emb,
                             float* __restrict__ zq) {
    int idx = blockIdx.x * blockDim.x + threadIdx.x;  // 64 * 262144
    int m = idx & (MPOS - 1);
    int c = idx >> 18;
    int n = m >> 10, hw = m & 1023;
    zq[((long)n * CDIM + c) * 1024 + hw] = emb[lat[m] * CDIM + c];
}

// ---------------------------------------------------------------------------
extern "C" void kernel_launch(void* const* d_in, const int* in_sizes, int n_in,
                              void* d_out, int out_size, void* d_ws, size_t ws_size,
                              hipStream_t stream) {
    const float* x      = (const float*)d_in[0];
    const float* enc_w1 = (const float*)d_in[1];
    const float* enc_b1 = (const float*)d_in[2];
    const float* bn1_g  = (const float*)d_in[3];
    const float* bn1_b  = (const float*)d_in[4];
    const float* enc_w2 = (const float*)d_in[5];
    const float* enc_b2 = (const float*)d_in[6];
    const float* bn2_g  = (const float*)d_in[7];
    const float* bn2_b  = (const float*)d_in[8];
    const float* enc_w3 = (const float*)d_in[9];
    const float* enc_b3 = (const float*)d_in[10];
    const float* bn3_g  = (const float*)d_in[11];
    const float* bn3_b  = (const float*)d_in[12];
    const float* emb    = (const float*)d_in[13];
    const float* dec_w1 = (const float*)d_in[14];
    const float* dec_b1 = (const float*)d_in[15];
    const float* bn4_g  = (const float*)d_in[16];
    const float* bn4_b  = (const float*)d_in[17];
    const float* dec_w2 = (const float*)d_in[18];
    const float* dec_b2 = (const float*)d_in[19];
    const float* bn5_g  = (const float*)d_in[20];
    const float* bn5_b  = (const float*)d_in[21];
    const float* dec_w3 = (const float*)d_in[22];
    const float* dec_b3 = (const float*)d_in[23];

    float* ws = (float*)d_ws;
    float* A    = ws;                 // 16,777,216 f  (conv1 out / later d2)
    float* B    = ws + 16777216;      //  8,388,608 f  (conv2 out / later d1)
    float* Cb   = ws + 25165824;      // 16,777,216 f  (conv3 raw)
    int*   lat  = (int*)(ws + 41943040);  // 262,144 i32
    float* part = ws + 42205184;      // <= 64*64*2 f partial sums
    float* scsh = ws + 42213376;      // 2*64 f fused scale/shift

    float* out  = (float*)d_out;
    float* ze   = out + 4194304;      // z_e_x
    float* zq   = ze + 16777216;      // z_q_x

    const int T = 256;

    // ---- encoder ----
    conv1_k<<<65536, T, 0, stream>>>(x, enc_w1, enc_b1, A);
    stats_partial_k<<<dim3(16, STATS_NB), T, 0, stream>>>(A, part, 16, 4096, NBATCH);
    stats_finalize_k<<<1, 64, 0, stream>>>(part, bn1_g, bn1_b, scsh, 16, 1.0f / (256.0f * 4096.0f));
    bn_apply_k<<<65536, T, 0, stream>>>(A, A, scsh, 16, 4096, 1);

    conv2_k<<<32768, T, 0, stream>>>(A, enc_w2, enc_b2, B);
    stats_partial_k<<<dim3(32, STATS_NB), T, 0, stream>>>(B, part, 32, 1024, NBATCH);
    stats_finalize_k<<<1, 64, 0, stream>>>(part, bn2_g, bn2_b, scsh, 32, 1.0f / (256.0f * 1024.0f));
    bn_apply_k<<<32768, T, 0, stream>>>(B, B, scsh, 32, 1024, 1);

    conv1x1_k<<<65536, T, 0, stream>>>(B, enc_w3, enc_b3, Cb, 32, 64, 1024);
    stats_partial_k<<<dim3(64, STATS_NB), T, 0, stream>>>(Cb, part, 64, 1024, NBATCH);
    stats_finalize_k<<<1, 64, 0, stream>>>(part, bn3_g, bn3_b, scsh, 64, 1.0f / (256.0f * 1024.0f));
    bn_apply_k<<<65536, T, 0, stream>>>(Cb, ze, scsh, 64, 1024, 0);   // z_e_x -> d_out

    // ---- VQ (WMMA f32 16x16x4 GEMM + argmin) ----
    size_t shmem = (size_t)(KCODES * CDIM + KCODES) * sizeof(float);  // 133,120 B
    vq_argmin_k<<<MPOS / (16 * VQ_WAVES), 32 * VQ_WAVES, shmem, stream>>>(ze, emb, lat);
    scatter_zq_k<<<65536, T, 0, stream>>>(lat, emb, zq);              // z_q_x -> d_out

    // ---- decoder ----
    conv1x1_k<<<32768, T, 0, stream>>>(zq, dec_w1, dec_b1, B, 64, 32, 1024);
    stats_partial_k<<<dim3(32, STATS_NB), T, 0, stream>>>(B, part, 32, 1024, NBATCH);
    stats_finalize_k<<<1, 64, 0, stream>>>(part, bn4_g, bn4_b, scsh, 32, 1.0f / (256.0f * 1024.0f));
    bn_apply_k<<<32768, T, 0, stream>>>(B, B, scsh, 32, 1024, 1);

    convT_k<<<65536, T, 0, stream>>>(B, dec_w2, dec_b2, A, 32, 16, 32, 64, 0);
    stats_partial_k<<<dim3(16, STATS_NB), T, 0, stream>>>(A, part, 16, 4096, NBATCH);
    stats_finalize_k<<<1, 64, 0, stream>>>(part, bn5_g, bn5_b, scsh, 16, 1.0f / (256.0f * 4096.0f));
    bn_apply_k<<<65536, T, 0, stream>>>(A, A, scsh, 16, 4096, 1);

    convT_k<<<16384, T, 0, stream>>>(A, dec_w3, dec_b3, out, 16, 1, 64, 128, 1); // x_tilde
}